// SimLinear_30399778521256
// MI455X (gfx1250) — compile-verified
//
#include <hip/hip_runtime.h>

typedef __attribute__((ext_vector_type(2))) float v2f;
typedef __attribute__((ext_vector_type(8))) float v8f;
typedef int v4i __attribute__((vector_size(16)));   // 16-byte payload type

#define IN_F    8192
#define OUT_F   8192
#define BATCH   50
#define M_PAD   64          // 50 batches padded to 4 M-tiles of 16
#define KC      64          // K chunk staged in LDS
#define NKC     (IN_F / KC) // 128 chunks
#define PAD     4           // LDS row pad (floats) -> conflict-free fragment reads
#define LSTR    (KC + PAD)  // 68 floats per LDS row
#define N_TILE  32          // out columns per workgroup (2 N-tiles of 16)
#define THREADS 256         // 8 waves (wave32)

#define AS1 __attribute__((address_space(1)))
#define AS3 __attribute__((address_space(3)))

// Async global->LDS copy (GLOBAL_LOAD_ASYNC_TO_LDS_B128, ASYNCcnt-tracked).
// No VGPR staging => no spills, natural overlap with WMMA compute.
__device__ __forceinline__ void async_b128(const float* g, float* l) {
#if __has_builtin(__builtin_amdgcn_global_load_async_to_lds_b128)
    __builtin_amdgcn_global_load_async_to_lds_b128(
        (AS1 v4i*)g, (AS3 v4i*)l, /*offset=*/0, /*cpol=*/0);
#else
    const unsigned lds_off = (unsigned)(unsigned long long)l; // low 32b = LDS addr
    asm volatile("global_load_async_to_lds_b128 %0, %1, off"
                 :: "v"(lds_off), "v"(g) : "memory");
#endif
}

__device__ __forceinline__ void wait_async0() {
#if __has_builtin(__builtin_amdgcn_s_wait_asynccnt)
    __builtin_amdgcn_s_wait_asynccnt(0);
#else
    asm volatile("s_wait_asynccnt 0x0" ::: "memory");
#endif
}

__global__ __launch_bounds__(THREADS)
void simlinear_wmma_f32(const float* __restrict__ x,
                        const float* __restrict__ w,
                        const float* __restrict__ bias,
                        float* __restrict__ out)
{
    __shared__ float xs[2][M_PAD * LSTR];   // x chunk, double buffered
    __shared__ float wsh[2][N_TILE * LSTR]; // W chunk, double buffered

    const int tid   = threadIdx.x;
    const int lane  = tid & 31;
    const int wave  = tid >> 5;
    const int mt    = wave >> 1;           // 0..3  : batch tile
    const int nt    = wave & 1;            // 0..1  : out-col tile
    const int obase = blockIdx.x * N_TILE;

    const int half  = lane >> 4;           // 0: K=0,1   1: K=2,3 (f32 WMMA layout)
    const int lm    = lane & 15;           // row within 16

    // Zero the padded batch rows (50..63) once, in both buffers. Async staging
    // below never writes them, so they stay zero for the whole K loop.
    for (int i = tid; i < 2 * (M_PAD - BATCH) * LSTR; i += THREADS) {
        const int b = i / ((M_PAD - BATCH) * LSTR);
        const int r = i % ((M_PAD - BATCH) * LSTR);
        xs[b][BATCH * LSTR + r] = 0.0f;
    }

    // Stage one K-chunk into LDS buffer `buf` with async b128 copies.
    auto stage = [&](int buf, int k0) {
        // x chunk: 50 rows x 64 floats, 16B-aligned LDS rows (stride 68*4=272B)
        #pragma unroll
        for (int t = 0; t < 4; ++t) {
            const int id  = tid + t * THREADS;     // 0..1023
            const int row = id >> 4;               // 16 float4 per row
            const int c4  = id & 15;
            if (row < BATCH)
                async_b128(x + row * IN_F + k0 + c4 * 4,
                           &xs[buf][row * LSTR + c4 * 4]);
        }
        // W chunk: 32 rows x 64 floats, fully coalesced along K
        #pragma unroll
        for (int t = 0; t < 2; ++t) {
            const int id  = tid + t * THREADS;     // 0..511
            const int row = id >> 4;
            const int c4  = id & 15;
            async_b128(w + (obase + row) * IN_F + k0 + c4 * 4,
                       &wsh[buf][row * LSTR + c4 * 4]);
        }
    };

    v8f acc = {};  // 16x16 f32 accumulator tile (8 VGPRs)

    stage(0, 0);
    wait_async0();
    __syncthreads();

    for (int kc = 0; kc < NKC; ++kc) {
        const int buf = kc & 1;
        // Prefetch next chunk into the other buffer while we compute this one.
        if (kc + 1 < NKC) stage(buf ^ 1, (kc + 1) * KC);

        // A fragment: lane lm -> batch row (mt*16+lm), K pair selected by half
        // B fragment: lane lm -> out col  (nt*16+lm), K pair selected by half
        const float* xrow = &xs [buf][(mt * 16 + lm) * LSTR + half * 2];
        const float* wrow = &wsh[buf][(nt * 16 + lm) * LSTR + half * 2];

        #pragma unroll
        for (int kk = 0; kk < KC; kk += 4) {
            v2f a = *(const v2f*)(xrow + kk);   // ds_load_b64, bank-conflict-free
            v2f b = *(const v2f*)(wrow + kk);
            acc = __builtin_amdgcn_wmma_f32_16x16x4_f32(
                false, a, false, b, (short)0, acc, false, false);
        }

        wait_async0();     // our async writes for chunk kc+1 landed in LDS
        __syncthreads();   // everyone's writes visible; buf also safe to refill
    }

    // Epilogue: D layout -> VGPR r holds M=r (lanes 0-15) / M=r+8 (lanes 16-31),
    // N = lane%16. Add bias, guard padded batches.
    const int oc = obase + nt * 16 + lm;
    const float bv = bias[oc];
    #pragma unroll
    for (int r = 0; r < 8; ++r) {
        const int bb = mt * 16 + r + half * 8;
        if (bb < BATCH)
            out[bb * OUT_F + oc] = acc[r] + bv;
    }
}

extern "C" void kernel_launch(void* const* d_in, const int* in_sizes, int n_in,
                              void* d_out, int out_size, void* d_ws, size_t ws_size,
                              hipStream_t stream) {
    const float* x  = (const float*)d_in[0];
    const float* w  = (const float*)d_in[1];
    const float* b  = (const float*)d_in[2];
    float* out      = (float*)d_out;

    dim3 grid(OUT_F / N_TILE);   // 256 workgroups
    dim3 block(THREADS);         // 256 threads = 8 wave32
    simlinear_wmma_f32<<<grid, block, 0, stream>>>(x, w, b, out);
}